// CRF_63204738728060
// MI455X (gfx1250) — compile-verified
//
#include <hip/hip_runtime.h>

// CRF log-likelihood for MI455X (gfx1250, wave32).
//
// Forward scan reformulated in the exp domain: per step,
//   alpha'[i][b] = log( (M @ E)[i][b] ) + colmax_b + rowmax_i + logit[i][b]
// with M = exp(T - rowmax) constant in VGPRs, E = exp(alpha - colmax).
// One wave per 16-batch tile (single-wave WG => barriers are free, DS pipe
// in-order per wave), alpha held in registers, 64x V_WMMA_F32_16X16X4_F32
// per step with dual accumulators, logit tile streamed by the TDM
// (tensor_load_to_lds, double-buffered, TENSORcnt-synchronized).

#define NUM_LABELS 51
#define LS 53          // LABEL_SIZE
#define START_LBL 51
#define END_LBL 52
#define BATCH 256
#define SEQ 2048
#define BT 16          // batch tile (= WMMA N)
#define PEM 68         // Em pitch (even for b64 loads, 4-bank lane stride)

typedef __attribute__((ext_vector_type(2))) float v2f;
typedef __attribute__((ext_vector_type(8))) float v8f;
typedef __attribute__((ext_vector_type(4))) unsigned int u32x4;
typedef __attribute__((ext_vector_type(8))) int i32x8;
typedef __attribute__((ext_vector_type(4))) int i32x4;

#if __has_builtin(__builtin_amdgcn_tensor_load_to_lds) && \
    __has_builtin(__builtin_amdgcn_s_wait_tensorcnt)
#define HAVE_TDM 1
#define TILE_WAIT(n) __builtin_amdgcn_s_wait_tensorcnt(n)
#else
#define HAVE_TDM 0
#define TILE_WAIT(n)
#endif

// ---------------------------------------------------------------------------
// Kernel 1: gold score = unary + binary along the gold path. One wave / batch.
// ---------------------------------------------------------------------------
__global__ __launch_bounds__(32)
void crf_gold_kernel(const float* __restrict__ logits,
                     const int* __restrict__ labels,
                     const int* __restrict__ lens,
                     const float* __restrict__ T,
                     float* __restrict__ gold) {
  const int b = blockIdx.x;
  const int lane = threadIdx.x;
  const int len = lens[b];
  const int* lab = labels + (size_t)b * SEQ;
  const float* lg = logits + (size_t)b * SEQ * NUM_LABELS;

  float acc = 0.0f;
  for (int t = lane; t < len; t += 32)                    // unary
    acc += lg[(size_t)t * NUM_LABELS + lab[t]];
  for (int k = lane; k <= len; k += 32) {                 // binary (to,from)
    const int fr = (k == 0) ? START_LBL : lab[k - 1];
    const int to = (k == len) ? END_LBL : lab[k];
    acc += T[to * LS + fr];
  }
  #pragma unroll
  for (int off = 16; off > 0; off >>= 1)
    acc += __shfl_xor(acc, off, 32);
  if (lane == 0) gold[b] = acc;
}

// ---------------------------------------------------------------------------
// TDM: load a (16 x 51 f32) logit tile (row stride SEQ*51) into LDS.
// D# built per CDNA5 ISA ch.8: group0 {count,lds_addr,global_addr,type=2},
// group1 {data_size=4B, tensor_dim0/1, tile_dim0/1, dim0_stride}.
// clang-23 toolchain: 6-arg builtin (g0, g1, g2, g3, g4, cpol).
// ---------------------------------------------------------------------------
__device__ __forceinline__ void tile_load(const float* gptr, float* dst) {
#if HAVE_TDM
  const unsigned long long ga = (unsigned long long)(uintptr_t)gptr;
  const unsigned ldsoff = (unsigned)(uintptr_t)dst;      // flat addr[31:0] = LDS offset
  u32x4 g0;
  g0.x = 1u;                                             // count=1, user mode
  g0.y = ldsoff;                                         // lds_addr
  g0.z = (unsigned)(ga & 0xFFFFFFFFu);                   // global_addr[31:0]
  g0.w = (unsigned)((ga >> 32) & 0x1FFFFFFu) | (2u << 30); // addr[56:32] | type=2
  const unsigned td0 = NUM_LABELS, td1 = BT;             // tensor dims
  const unsigned tl0 = NUM_LABELS, tl1 = BT;             // tile dims
  const unsigned long long s0 = (unsigned long long)SEQ * NUM_LABELS; // dim0 stride
  i32x8 g1;
  g1[0] = (int)(2u << 16);                               // data_size=4B; no mask/flags
  g1[1] = (int)((td0 & 0xFFFFu) << 16);                  // tensor_dim0[15:0]
  g1[2] = (int)((td0 >> 16) | ((td1 & 0xFFFFu) << 16));  // dim0[31:16] | dim1[15:0]
  g1[3] = (int)((td1 >> 16) | (tl0 << 16));              // dim1[31:16] | tile_dim0
  g1[4] = (int)(tl1);                                    // tile_dim1 | tile_dim2=0
  g1[5] = (int)(s0 & 0xFFFFFFFFu);                       // dim0_stride[31:0]
  g1[6] = (int)((s0 >> 32) & 0xFFFFu);                   // stride[47:32] | dim1_stride lo
  g1[7] = 0;
  const i32x4 z4 = {0, 0, 0, 0};                         // groups 2/3 unused (2D)
  const i32x8 z8 = {0, 0, 0, 0, 0, 0, 0, 0};             // extra group (unused)
  __builtin_amdgcn_tensor_load_to_lds(g0, g1, z4, z4, z8, 0);
#else
  for (int idx = threadIdx.x; idx < BT * NUM_LABELS; idx += 32) {
    const int b = idx / NUM_LABELS, j = idx - b * NUM_LABELS;
    dst[idx] = gptr[(size_t)b * SEQ * NUM_LABELS + j];
  }
#endif
}

// ---------------------------------------------------------------------------
// Kernel 2: forward scan, one wave per 16-batch tile. Lane owns one batch
// column (2 lanes/column) and 32 alpha rows in registers.
// ---------------------------------------------------------------------------
__global__ __launch_bounds__(32)
void crf_forward_kernel(const float* __restrict__ logits,
                        const int* __restrict__ lens,
                        const float* __restrict__ T,
                        const float* __restrict__ gold,
                        float* __restrict__ out) {
  __shared__ float Em[BT * PEM];            // E, column-major [batch][label]
  __shared__ float lgt[2][BT * NUM_LABELS]; // TDM double buffer [batch][label]
  __shared__ float rsh_lds[64];             // per-row exp shift

  const int lane  = threadIdx.x;
  const int rowlo = lane & 15;
  const int khalf = (lane < 16) ? 0 : 2;    // A/B K-halves per f32 16x4 layout
  const int off   = (lane < 16) ? 0 : 8;    // D-row half per C/D layout
  const int colb  = rowlo;                  // this lane's batch column
  const int b0    = blockIdx.x * BT;

  // rowshift[i] = max_j T[i][j] (i < LS), else 0 (pad rows are all-zero in M)
  for (int row = lane; row < 64; row += 32) {
    float m = 0.0f;
    if (row < LS) {
      m = -1e30f;
      for (int j = 0; j < LS; ++j) m = fmaxf(m, T[row * LS + j]);
    }
    rsh_lds[row] = m;
  }
  __syncthreads();  // single-wave WG: ~free; orders rsh_lds for reads below

  // A operands: M[i][j] = exp(T[i][j] - rowshift[i]), zero padding.
  // f32 16x4 A: lanes 0-15 hold K={4k,4k+1}, lanes 16-31 hold K={4k+2,4k+3}.
  v2f a[4][16];
  #pragma unroll
  for (int m = 0; m < 4; ++m) {
    const int ar = 16 * m + rowlo;
    const float rs = rsh_lds[ar];
    #pragma unroll
    for (int k = 0; k < 16; ++k) {
      const int j0 = 4 * k + khalf;
      a[m][k].x = (ar < LS && (j0    ) < LS) ? __expf(T[ar * LS + j0    ] - rs) : 0.0f;
      a[m][k].y = (ar < LS && (j0 + 1) < LS) ? __expf(T[ar * LS + j0 + 1] - rs) : 0.0f;
    }
  }

  // alpha in registers: al[m][r] = alpha[16m + off + r][colb]
  float al[4][8];
  #pragma unroll
  for (int m = 0; m < 4; ++m)
    #pragma unroll
    for (int r = 0; r < 8; ++r) {
      const int row = 16 * m + off + r;
      al[m][r] = (row == START_LBL) ? 0.0f : ((row < LS) ? -100.0f : -1e30f);
    }

  const int len = lens[b0 + colb];
  const float* gtile = logits + (size_t)b0 * SEQ * NUM_LABELS;

  tile_load(gtile, &lgt[0][0]);             // async DMA for t = 0

  for (int t = 0; t < SEQ; ++t) {
    // Issue DMA for t+1 (other buffer). dscnt guard: prior ds_loads on that
    // buffer (step t-1) must be out of the DS pipe before TDM overwrites it.
    if (t + 1 < SEQ) {
      asm volatile("s_wait_dscnt 0" ::: "memory");
      tile_load(gtile + (size_t)(t + 1) * NUM_LABELS, &lgt[(t + 1) & 1][0]);
    }

    // column max of alpha: register reduce + cross-half shuffle
    float cm = -1e30f;
    #pragma unroll
    for (int m = 0; m < 4; ++m)
      #pragma unroll
      for (int r = 0; r < 8; ++r) cm = fmaxf(cm, al[m][r]);
    cm = fmaxf(cm, __shfl_xor(cm, 16, 32));

    // E = exp(alpha - colmax) -> LDS (b128-friendly runs, conflict-free)
    #pragma unroll
    for (int m = 0; m < 4; ++m) {
      float* ep = &Em[colb * PEM + 16 * m + off];
      #pragma unroll
      for (int r = 0; r < 8; ++r) ep[r] = __expf(al[m][r] - cm);
    }

    // B operands (shared by all 4 M-tiles): b64 loads, lane stride 4 banks
    v2f bb[16];
    #pragma unroll
    for (int k = 0; k < 16; ++k)
      bb[k] = *(const v2f*)&Em[rowlo * PEM + 4 * k + khalf];

    // wait for this step's logit tile (oldest of <=2 outstanding transfers)
    if (t + 1 < SEQ) { TILE_WAIT(1); } else { TILE_WAIT(0); }
    const float* lrow = &lgt[t & 1][colb * NUM_LABELS];
    const bool keep = (t < len);

    // 4 tiles x 16 K-steps, dual accumulators to halve the WMMA RAW chain
    #pragma unroll
    for (int m = 0; m < 4; ++m) {
      v8f c0 = {}, c1 = {};
      #pragma unroll
      for (int k = 0; k < 16; k += 2) {
        c0 = __builtin_amdgcn_wmma_f32_16x16x4_f32(
            false, a[m][k],     false, bb[k],     (short)0, c0, false, false);
        c1 = __builtin_amdgcn_wmma_f32_16x16x4_f32(
            false, a[m][k + 1], false, bb[k + 1], (short)0, c1, false, false);
      }
      #pragma unroll
      for (int r = 0; r < 8; ++r) {
        const int row = 16 * m + off + r;
        const float lv = (row < NUM_LABELS) ? lrow[row] : -100.0f;
        const float v  = __logf(c0[r] + c1[r]) + cm + rsh_lds[row] + lv;
        al[m][r] = keep ? v : al[m][r];
      }
    }
  }

  // norm = logsumexp_j(alpha[j] + T[END][j]) over real labels; out = gold-norm
  float mx = -1e30f;
  #pragma unroll
  for (int m = 0; m < 4; ++m)
    #pragma unroll
    for (int r = 0; r < 8; ++r) {
      const int row = 16 * m + off + r;
      if (row < LS) mx = fmaxf(mx, al[m][r] + T[END_LBL * LS + row]);
    }
  mx = fmaxf(mx, __shfl_xor(mx, 16, 32));
  float s = 0.0f;
  #pragma unroll
  for (int m = 0; m < 4; ++m)
    #pragma unroll
    for (int r = 0; r < 8; ++r) {
      const int row = 16 * m + off + r;
      if (row < LS) s += __expf(al[m][r] + T[END_LBL * LS + row] - mx);
    }
  s += __shfl_xor(s, 16, 32);
  const float norm = __logf(s) + mx;
  if (lane < 16) out[b0 + colb] = gold[b0 + colb] - norm;
}

// ---------------------------------------------------------------------------
extern "C" void kernel_launch(void* const* d_in, const int* in_sizes, int n_in,
                              void* d_out, int out_size, void* d_ws, size_t ws_size,
                              hipStream_t stream) {
  const float* logits = (const float*)d_in[0];   // (256, 2048, 51) f32
  const int*   labels = (const int*)d_in[1];     // (256, 2048) int
  const int*   lens   = (const int*)d_in[2];     // (256,) int
  const float* T      = (const float*)d_in[3];   // (53, 53) f32
  float* out  = (float*)d_out;                   // (256,) f32
  float* gold = (float*)d_ws;                    // 256 floats scratch

  crf_gold_kernel<<<BATCH, 32, 0, stream>>>(logits, labels, lens, T, gold);
  crf_forward_kernel<<<BATCH / BT, 32, 0, stream>>>(logits, lens, T, gold, out);
}